// BacklashNet_86870008529014
// MI455X (gfx1250) — compile-verified
//
#include <hip/hip_runtime.h>

// Problem geometry (fixed by the reference).
#define T_LEN   8192
#define B_LEN   2048
#define ROWS    32            // rows (batch elements) per workgroup == lanes per wave
#define CHUNK   128           // time-steps staged per tile
#define STRIDE  132           // floats per LDS row: 128 + 4 pad (16B-aligned rows, bank rotation)
#define NCHUNK  (T_LEN / CHUNK)   // 64

// Builtin takes pointers to 16-byte int vectors (per hipcc diagnostic):
//   param0: AS1 (__device__) v4i*   param1: AS3 (LDS) v4i*
typedef int v4i __attribute__((vector_size(16)));
typedef __attribute__((address_space(1))) v4i g_v4i;   // global
typedef __attribute__((address_space(3))) v4i l_v4i;   // LDS

#if defined(__HIP_DEVICE_COMPILE__) && \
    __has_builtin(__builtin_amdgcn_global_load_async_to_lds_b128) && \
    __has_builtin(__builtin_amdgcn_s_wait_asynccnt)
#define USE_ASYNC 1
#else
#define USE_ASYNC 0
#endif

// One backlash step. Dependent chain on p: 2 parallel v_cmp -> 2 parallel
// v_cndmask -> 1 v_cndmask (depth 3). lo/up/s depend only on x and pipeline
// ahead of the chain.
__device__ __forceinline__ float bl_step(float p, float xt,
                                         float m_lo, float mlc,
                                         float m_up, float muc) {
  float lo = m_lo * xt + mlc;   // m_lo*(x + c_lo)
  float up = m_up * xt + muc;   // m_up*(x + c_up)
  float s  = lo + up;
  bool f1 = (lo <= p);          // s1 <= 0
  bool f2 = (p  <= up);         // s2 <= 0
  float a = f2 ? s  : lo;       // taken when f1
  float b = f2 ? up : p;        // taken when !f1
  return f1 ? a : b;
}

__global__ __launch_bounds__(ROWS)
void backlash_scan(const float* __restrict__ x,
                   const float* __restrict__ prev0,
                   const float* __restrict__ w,
                   float* __restrict__ out) {
  // Double-buffered input tile: [2][32 rows][132 floats] = 33 KiB LDS.
  __shared__ float sb[2][ROWS * STRIDE];

  const int lane = threadIdx.x;           // 0..31, one row per lane
  const int row0 = blockIdx.x * ROWS;
  const int row  = row0 + lane;

  const float m_lo = w[0];
  const float m_up = w[1];
  const float c_lo = w[2];
  const float c_up = w[3];
  const float mlc  = m_lo * c_lo;
  const float muc  = m_up * c_up;

  float p = prev0[row];

  // Stage one 32x128 tile (rows row0..row0+31, cols t0..t0+127) into sb[kbuf].
  // Instruction j copies row j: lanes cover 128 consecutive floats (512B,
  // perfectly coalesced), landing at per-lane LDS addresses in the padded
  // layout. 32 async b128 instructions per tile.
  auto issue_tile = [&](int kbuf, int t0) {
    const float* g0 = x + (size_t)row0 * T_LEN + t0 + 4 * lane;
    float*       l0 = &sb[kbuf][4 * lane];
#pragma unroll
    for (int j = 0; j < ROWS; ++j) {
      const float* g = g0 + (size_t)j * T_LEN;
      float*       l = l0 + j * STRIDE;
#if USE_ASYNC
      __builtin_amdgcn_global_load_async_to_lds_b128(
          (g_v4i*)g, (l_v4i*)l, /*offset=*/0, /*cpol=*/0);
#else
      *(float4*)l = *(const float4*)g;
#endif
    }
  };

  issue_tile(0, 0);   // prefetch tile 0

  float* __restrict__ orow = out + (size_t)row * T_LEN;

  for (int k = 0; k < NCHUNK; ++k) {
    const int cur = k & 1;

    if (k + 1 < NCHUNK) {
      issue_tile(cur ^ 1, (k + 1) * CHUNK);   // overlap next tile's DMA
#if USE_ASYNC
      // Async loads complete in order: <=32 outstanding means tile k is
      // fully resident while tile k+1's 32 loads remain in flight.
      __builtin_amdgcn_s_wait_asynccnt(ROWS);
#else
      __syncthreads();
#endif
    } else {
#if USE_ASYNC
      __builtin_amdgcn_s_wait_asynccnt(0);
#else
      __syncthreads();
#endif
    }
    asm volatile("" ::: "memory");   // keep LDS reads below the wait

    // Lane l scans row l of the tile. Padded stride (132 % 64 == 4) rotates
    // banks across lanes; row starts stay 16B-aligned for ds_load_b128.
    const float* __restrict__ myrow = &sb[cur][lane * STRIDE];
    float* __restrict__ oc = orow + k * CHUNK;

#pragma unroll 8
    for (int c = 0; c < CHUNK; c += 4) {
      float4 xv = *(const float4*)(myrow + c);
      float4 y;
      p = bl_step(p, xv.x, m_lo, mlc, m_up, muc); y.x = p;
      p = bl_step(p, xv.y, m_lo, mlc, m_up, muc); y.y = p;
      p = bl_step(p, xv.z, m_lo, mlc, m_up, muc); y.z = p;
      p = bl_step(p, xv.w, m_lo, mlc, m_up, muc); y.w = p;
      *(float4*)(oc + c) = y;        // 16B store, off the dependent chain
    }
  }
}

extern "C" void kernel_launch(void* const* d_in, const int* in_sizes, int n_in,
                              void* d_out, int out_size, void* d_ws, size_t ws_size,
                              hipStream_t stream) {
  const float* x     = (const float*)d_in[0];   // (B, T, 1) f32
  const float* prev0 = (const float*)d_in[1];   // (B, 1, 1) f32
  const float* w     = (const float*)d_in[2];   // (4,) f32
  float*       out   = (float*)d_out;           // (B, T, 1) f32

  dim3 grid(B_LEN / ROWS);   // 64 workgroups
  dim3 block(ROWS);          // 32 threads = 1 wave32 each
  backlash_scan<<<grid, block, 0, stream>>>(x, prev0, w, out);
}